// CrossHatchPowerFractal_10857677325042
// MI455X (gfx1250) — compile-verified
//
#include <hip/hip_runtime.h>

#define WI 1024
#define HI 1024
#define NPIX (WI * HI)
#define OCTAVES 12
#define NCOL 24   /* 8 batches * 3 channels */

typedef __attribute__((ext_vector_type(2))) float v2f;
typedef __attribute__((ext_vector_type(8))) float v8f;

// ---------------------------------------------------------------------------
// Kernel 1: one block of 1024 threads computes the 1D fractal profile
//   axn[x] = (A(x) - Amin) * 7.5 / (Amax - Amin)      (so idx = rint(axn[x]+axn[y]))
// and initializes the global min/max atomic slots.
// ---------------------------------------------------------------------------
__device__ inline float blk_min(float v, float* red, int t) {
  red[t] = v; __syncthreads();
  for (int s = 512; s > 0; s >>= 1) {
    if (t < s) red[t] = fminf(red[t], red[t + s]);
    __syncthreads();
  }
  float r = red[0]; __syncthreads();
  return r;
}
__device__ inline float blk_max(float v, float* red, int t) {
  red[t] = v; __syncthreads();
  for (int s = 512; s > 0; s >>= 1) {
    if (t < s) red[t] = fmaxf(red[t], red[t + s]);
    __syncthreads();
  }
  float r = red[0]; __syncthreads();
  return r;
}

__global__ __launch_bounds__(1024) void k_profile(float* __restrict__ ws) {
  __shared__ float red[1024];
  const int i = threadIdx.x;
  // linspace(0,1,1024): i*step with endpoint forced to 1.0 (f32, like the ref)
  const float xg = (i == 1023) ? 1.0f : (float)i * (1.0f / 1023.0f);
  float A = 0.0f;
  double pw = 1.0;                       // persistence^o, exact in double
  for (int o = 0; o < OCTAVES; ++o) {
    // f*pi computed in double then rounded to f32 (matches Python-float -> f32)
    const float fpi = (float)((double)(320 << o) * 3.14159265358979323846);
    const float s = sinf(xg * fpi);      // f32 product, f32 sin, like the ref
    const float mn = blk_min(s, red, i);
    const float mx = blk_max(s, red, i);
    A += ((s - mn) / (mx - mn)) * (float)pw;
    pw *= 1.5;
  }
  const float Amn = blk_min(A, red, i);
  const float Amx = blk_max(A, red, i);
  ws[i] = (A - Amn) * (7.5f / (Amx - Amn));
  if (i == 0) {
    unsigned* mm = (unsigned*)(ws + WI);
    mm[0] = 0xFFFFFFFFu;   // min slot (uint order == float order for vals >= 0)
    mm[1] = 0u;            // max slot
  }
}

// ---------------------------------------------------------------------------
// Kernel 2: WMMA blur. One wave = 16 consecutive y pixels at one x row.
//   blurred[pix, n] = (1/25) * counts[pix, k] @ P[k, n],  n = b*3+c (24 cols)
// counts = 16-bin histogram of idx over the 5x5 zero-padded window.
// pass==0: reduce global min/max into mm slots.
// pass==1: write normalized output (B, W, H, 3).
// A-fragment (f32 16x4): VGPR v, lane-half h  ->  K = 4j + v + 2h
// B-fragment (f32 4x16): VGPR v, lane-half h  ->  K = 4j + v + 2h, N = lane&15
// C/D (f32 16x16):       VGPR r, lane-half h  ->  M = r + 8h,      N = lane&15
// ---------------------------------------------------------------------------
__global__ __launch_bounds__(128) void k_blur(const float* __restrict__ colors,
                                              const float* __restrict__ axn,
                                              unsigned* __restrict__ mm,
                                              float* __restrict__ out,
                                              int pass) {
  __shared__ float smn[128];
  __shared__ float smx[128];

  const int t    = threadIdx.x;
  const int lane = t & 31;
  const int wave = t >> 5;
  const int p    = lane & 15;   // pixel-in-tile (A rows / C columns index)
  const int h    = lane >> 4;   // lane half
  const int x    = blockIdx.y;                 // 0..1023
  const int y0   = (blockIdx.x * 4 + wave) * 16;
  const int y    = y0 + p;

  // 5-tap neighborhoods of the 1D profile (1e30 sentinel => rint is huge,
  // never matches a bin => zero-padded window, divisor stays 25)
  float axv[5], ayv[5];
#pragma unroll
  for (int d = 0; d < 5; ++d) {
    const int xx = x + d - 2;
    axv[d] = (xx >= 0 && xx < WI) ? axn[xx] : 1.0e30f;
    const int yy = y + d - 2;
    ayv[d] = (yy >= 0 && yy < HI) ? axn[yy] : 1.0e30f;
  }

  // Per-lane slice of the window histogram: k = 4j + v + 2h
  float kf[4][2], cnt[4][2];
#pragma unroll
  for (int j = 0; j < 4; ++j)
#pragma unroll
    for (int v = 0; v < 2; ++v) { kf[j][v] = (float)(4 * j + v + 2 * h); cnt[j][v] = 0.0f; }

#pragma unroll
  for (int dx = 0; dx < 5; ++dx) {
#pragma unroll
    for (int dy = 0; dy < 5; ++dy) {
      const float fi = rintf(axv[dx] + ayv[dy]);   // exact small integer in [0,15]
#pragma unroll
      for (int j = 0; j < 4; ++j)
#pragma unroll
        for (int v = 0; v < 2; ++v)
          cnt[j][v] += (fi == kf[j][v]) ? 1.0f : 0.0f;
    }
  }

  // Palette B-fragments. P[k][n] = colors[((n/3)*16 + k)*3 + (n%3)], n = b*3+c
  const int n0 = p;        // N-tile 0 column, always valid (< 24)
  const int n1 = 16 + p;   // N-tile 1 column, valid only for p < 8
  const bool v1 = (n1 < NCOL);
  v2f b0[4], b1[4];
#pragma unroll
  for (int j = 0; j < 4; ++j)
#pragma unroll
    for (int v = 0; v < 2; ++v) {
      const int k = 4 * j + v + 2 * h;
      b0[j][v] = colors[((n0 / 3) * 16 + k) * 3 + (n0 % 3)];
      b1[j][v] = v1 ? colors[((n1 / 3) * 16 + k) * 3 + (n1 % 3)] : 0.0f;
    }

  v8f acc0 = {0.f, 0.f, 0.f, 0.f, 0.f, 0.f, 0.f, 0.f};
  v8f acc1 = {0.f, 0.f, 0.f, 0.f, 0.f, 0.f, 0.f, 0.f};
#pragma unroll
  for (int j = 0; j < 4; ++j) {
    v2f a; a[0] = cnt[j][0]; a[1] = cnt[j][1];
    acc0 = __builtin_amdgcn_wmma_f32_16x16x4_f32(false, a, false, b0[j],
                                                 (short)0, acc0, false, false);
    acc1 = __builtin_amdgcn_wmma_f32_16x16x4_f32(false, a, false, b1[j],
                                                 (short)0, acc1, false, false);
  }

  const float inv25 = 1.0f / 25.0f;
  if (pass == 0) {
    float lmn = 1.0e30f, lmx = -1.0e30f;
#pragma unroll
    for (int r = 0; r < 8; ++r) {
      const float a = acc0[r] * inv25;
      lmn = fminf(lmn, a); lmx = fmaxf(lmx, a);
      if (v1) {
        const float b = acc1[r] * inv25;
        lmn = fminf(lmn, b); lmx = fmaxf(lmx, b);
      }
    }
    smn[t] = lmn; smx[t] = lmx; __syncthreads();
    for (int s = 64; s > 0; s >>= 1) {
      if (t < s) { smn[t] = fminf(smn[t], smn[t + s]); smx[t] = fmaxf(smx[t], smx[t + s]); }
      __syncthreads();
    }
    if (t == 0) {
      atomicMin(&mm[0], __float_as_uint(smn[0]));   // all values >= 0
      atomicMax(&mm[1], __float_as_uint(smx[0]));
    }
  } else {
    const float mn = __uint_as_float(mm[0]);
    const float mx = __uint_as_float(mm[1]);
    const float sc = 1.0f / (mx - mn);   // CLAMP range is [0,1]; bright/contrast identity
    const int b0i = n0 / 3, c0i = n0 % 3;
    const int b1i = n1 / 3, c1i = n1 % 3;
#pragma unroll
    for (int r = 0; r < 8; ++r) {
      const int pix = x * HI + (y0 + 8 * h + r);   // out[b][x][y][c]
      out[(size_t)b0i * (3 * NPIX) + (size_t)pix * 3 + c0i] = (acc0[r] * inv25 - mn) * sc;
      if (v1)
        out[(size_t)b1i * (3 * NPIX) + (size_t)pix * 3 + c1i] = (acc1[r] * inv25 - mn) * sc;
    }
  }
}

// ---------------------------------------------------------------------------
extern "C" void kernel_launch(void* const* d_in, const int* in_sizes, int n_in,
                              void* d_out, int out_size, void* d_ws, size_t ws_size,
                              hipStream_t stream) {
  (void)in_sizes; (void)n_in; (void)out_size; (void)ws_size;
  const float* colors = (const float*)d_in[0];     // (8, 16, 3) f32
  float* out = (float*)d_out;                      // (8, 1024, 1024, 3) f32
  float* ws  = (float*)d_ws;                       // axn[1024] + 2 minmax slots
  unsigned* mm = (unsigned*)(ws + WI);

  k_profile<<<dim3(1), dim3(1024), 0, stream>>>(ws);

  const dim3 grid(16, 1024);   // blockIdx.x*4+wave = y-tile (64), blockIdx.y = x row
  const dim3 block(128);       // 4 waves per block
  k_blur<<<grid, block, 0, stream>>>(colors, ws, mm, out, 0);  // min/max pass
  k_blur<<<grid, block, 0, stream>>>(colors, ws, mm, out, 1);  // normalize+store
}